// HyFormerBlock_7327214207605
// MI455X (gfx1250) — compile-verified
//
#include <hip/hip_runtime.h>
#include <hip/hip_bf16.h>
#include <math.h>

typedef __attribute__((ext_vector_type(2))) float v2f;
typedef __attribute__((ext_vector_type(4))) float v4f;
typedef __attribute__((ext_vector_type(8))) float v8f;

#define Bsz 32
#define Ssz 3
#define Nsz 8
#define Msz 8
#define Lsz 512
#define Dsz 512
#define Hsz 1024
#define Tsz 16

#define TILE_M 64
#define TILE_N 64
#define TILE_K 32
#define LDA_S 36   // A LDS row stride (floats): 36*m mod 64 distinct over 16 rows, f4-alignable
#define LDB_S 72   // B LDS row stride: 2*72 mod 64 = 16 -> lane halves hit disjoint bank windows

// ---------------------------------------------------------------------------
// Staging: 64x32 A tile (512 float4) + 32x64 B tile (512 float4), 4 each per thread.
// ---------------------------------------------------------------------------
__device__ __forceinline__ void stage_tiles(const float* __restrict__ Ab,
                                            const float* __restrict__ Bb,
                                            float* __restrict__ As,
                                            float* __restrict__ Bs,
                                            int m0, int n0, int k0,
                                            int M, int N, int K, bool fullM)
{
    const int tid = threadIdx.x;
    if (fullM) {                       // block-uniform fast path: no exec masking
        #pragma unroll
        for (int i = 0; i < 4; ++i) {
            int idx = tid + 128 * i;
            int r = idx >> 3, kq = idx & 7;      // 8 float4 per 32-float row chunk
            *(v4f*)&As[r * LDA_S + kq * 4] =
                *(const v4f*)(Ab + (long long)(m0 + r) * K + k0 + kq * 4);
        }
    } else {
        #pragma unroll
        for (int i = 0; i < 4; ++i) {
            int idx = tid + 128 * i;
            int r = idx >> 3, kq = idx & 7;
            if (m0 + r < M)
                *(v4f*)&As[r * LDA_S + kq * 4] =
                    *(const v4f*)(Ab + (long long)(m0 + r) * K + k0 + kq * 4);
        }
    }
    #pragma unroll
    for (int i = 0; i < 4; ++i) {
        int idx = tid + 128 * i;
        int r = idx >> 4, nq = idx & 15;         // 16 float4 per 64-float row
        *(v4f*)&Bs[r * LDB_S + nq * 4] =
            *(const v4f*)(Bb + (long long)(k0 + r) * N + n0 + nq * 4);
    }
}

// ---------------------------------------------------------------------------
// Compute: 8 k-steps x 4 wmma (2x2 wave tile) on one staged 32-deep chunk.
// A frag (16x4 f32): lanes 0-15 K=kb,kb+1 / lanes 16-31 K=kb+2,kb+3.
// B frag (4x16):     VGPR r holds rows K=r and K=r+2 across lane halves.
// ---------------------------------------------------------------------------
__device__ __forceinline__ void compute_tiles(const float* __restrict__ As,
                                              const float* __restrict__ Bs,
                                              v8f acc[2][2],
                                              int wm, int wn, int lhalf, int l16)
{
    #pragma unroll
    for (int kb = 0; kb < TILE_K; kb += 4) {
        const int kk = kb + lhalf * 2;
        v2f afrag[2], bfrag[2];
        #pragma unroll
        for (int mi = 0; mi < 2; ++mi) {
            int row = wm * 32 + mi * 16 + l16;
            afrag[mi] = *(const v2f*)&As[row * LDA_S + kk];
        }
        #pragma unroll
        for (int ni = 0; ni < 2; ++ni) {
            int col = wn * 32 + ni * 16 + l16;
            bfrag[ni].x = Bs[kk * LDB_S + col];
            bfrag[ni].y = Bs[(kk + 1) * LDB_S + col];
        }
        #pragma unroll
        for (int mi = 0; mi < 2; ++mi)
            #pragma unroll
            for (int ni = 0; ni < 2; ++ni)
                acc[mi][ni] = __builtin_amdgcn_wmma_f32_16x16x4_f32(
                    false, afrag[mi], false, bfrag[ni],
                    (short)0, acc[mi][ni], false, false);
    }
}

// ---------------------------------------------------------------------------
// Generic batched tiled GEMM: C[z] = A[z] (MxK row-major) * B[z%bmod] (KxN) + bias[z%bmod]
// 128 threads = 4 waves (2x2), 32x32 per wave, double-buffered 32-deep LDS chunks.
// ---------------------------------------------------------------------------
__global__ __launch_bounds__(128)
void gemm_wmma_f32_kernel(const float* __restrict__ A, long long strideA,
                          const float* __restrict__ Bm, long long strideB, int bmod,
                          const float* __restrict__ bias, long long strideBias,
                          float* __restrict__ C, long long strideC,
                          int M, int N, int K)
{
    __shared__ float As[2][TILE_M * LDA_S];
    __shared__ float Bs[2][TILE_K * LDB_S];

    const int z = blockIdx.z;
    const float* Ab = A + (long long)z * strideA;
    const float* Bb = Bm + (long long)(z % bmod) * strideB;
    const float* biasb = bias ? (bias + (long long)(z % bmod) * strideBias) : nullptr;
    float* Cb = C + (long long)z * strideC;

    const int tid   = threadIdx.x;
    const int lane  = tid & 31;
    const int wid   = tid >> 5;      // 0..3
    const int wm    = wid >> 1;      // wave row 0..1
    const int wn    = wid & 1;       // wave col 0..1
    const int lhalf = lane >> 4;     // 0|1
    const int l16   = lane & 15;

    const int m0 = blockIdx.x * TILE_M;
    const int n0 = blockIdx.y * TILE_N;
    const bool fullM = (m0 + TILE_M) <= M;

    v8f acc[2][2] = {};

    const int nst = K / TILE_K;      // K is a multiple of 32 for every launch here
    stage_tiles(Ab, Bb, As[0], Bs[0], m0, n0, 0, M, N, K, fullM);

    int cur = 0;
    for (int t = 0; t < nst; ++t) {
        __syncthreads();             // staged chunk visible; prev buffer free to overwrite
        if (t + 1 < nst)
            stage_tiles(Ab, Bb, As[1 - cur], Bs[1 - cur],
                        m0, n0, (t + 1) * TILE_K, M, N, K, fullM);
        compute_tiles(As[cur], Bs[cur], acc, wm, wn, lhalf, l16);
        cur ^= 1;
    }

    // ---- store: D VGPR r -> row r (lanes 0-15) / r+8 (lanes 16-31), col = l16 ----
    #pragma unroll
    for (int mi = 0; mi < 2; ++mi) {
        #pragma unroll
        for (int ni = 0; ni < 2; ++ni) {
            int col = n0 + wn * 32 + ni * 16 + l16;
            #pragma unroll
            for (int r = 0; r < 8; ++r) {
                int row = m0 + wm * 32 + mi * 16 + r + 8 * lhalf;
                if (row < M) {
                    float v = acc[mi][ni][r];
                    if (biasb) v += biasb[col];
                    Cb[(long long)row * N + col] = v;
                }
            }
        }
    }
}

// ---------------------------------------------------------------------------
__device__ __forceinline__ float block_rms_inv(float v0, float v1, float* red)
{
    const int tid = threadIdx.x;
    red[tid] = v0 * v0 + v1 * v1;
    __syncthreads();
    for (int o = 128; o > 0; o >>= 1) {
        if (tid < o) red[tid] += red[tid + o];
        __syncthreads();
    }
    float inv = rsqrtf(red[0] * (1.0f / 512.0f) + 1e-6f);
    __syncthreads();
    return inv;
}

// rms_norm of 512-wide rows; weight row s = (row / wdiv) % S
__global__ __launch_bounds__(256)
void rmsnorm_rows_kernel(const float* __restrict__ X, const float* __restrict__ w,
                         float* __restrict__ Y, int wdiv)
{
    __shared__ float red[256];
    const long long row = blockIdx.x;
    const int tid = threadIdx.x;
    const float* x = X + row * Dsz;
    float* y = Y + row * Dsz;
    const int s = (int)((row / wdiv) % Ssz);
    const float* ws = w + (long long)s * Dsz;
    float v0 = x[tid], v1 = x[tid + 256];
    float inv = block_rms_inv(v0, v1, red);
    y[tid]       = v0 * inv * ws[tid];
    y[tid + 256] = v1 * inv * ws[tid + 256];
}

// ---------------------------------------------------------------------------
// Attention per (h, s, b): masked scaled scores, wave-per-row softmax, attn @ V.
// ---------------------------------------------------------------------------
__global__ __launch_bounds__(256)
void attention_kernel(const float* __restrict__ Qn, const float* __restrict__ Kbuf,
                      const float* __restrict__ Vbuf, const unsigned char* __restrict__ mask,
                      float* __restrict__ attn_out)
{
    const int h = blockIdx.x, s = blockIdx.y, b = blockIdx.z;
    const long long bs = (long long)b * Ssz + s;
    __shared__ float qh[Nsz * 64];
    __shared__ float sc[Nsz * Lsz];
    const int tid = threadIdx.x;

    for (int i = tid; i < Nsz * 64; i += 256) {
        int n = i >> 6, d = i & 63;
        qh[i] = Qn[(bs * Nsz + n) * Dsz + h * 64 + d];
    }
    __syncthreads();

    const float NEG = -3.402823466e38f;
    for (int l = tid; l < Lsz; l += 256) {
        const float* kp = Kbuf + (bs * Lsz + l) * Dsz + h * 64;
        float acc[Nsz] = {};
        for (int d = 0; d < 64; ++d) {
            float kv = kp[d];
            #pragma unroll
            for (int n = 0; n < Nsz; ++n) acc[n] += qh[n * 64 + d] * kv;
        }
        const bool valid = mask[bs * Lsz + l] != 0;
        #pragma unroll
        for (int n = 0; n < Nsz; ++n)
            sc[n * Lsz + l] = valid ? acc[n] * 0.125f : NEG;
    }
    __syncthreads();

    {   // wave w -> softmax of row w
        const int w = tid >> 5, lane = tid & 31;
        float mx = NEG;
        for (int l = lane; l < Lsz; l += 32) mx = fmaxf(mx, sc[w * Lsz + l]);
        for (int off = 16; off > 0; off >>= 1) mx = fmaxf(mx, __shfl_xor(mx, off));
        float sum = 0.0f;
        for (int l = lane; l < Lsz; l += 32) {
            float e = __expf(sc[w * Lsz + l] - mx);
            sc[w * Lsz + l] = e;
            sum += e;
        }
        for (int off = 16; off > 0; off >>= 1) sum += __shfl_xor(sum, off);
        float invs = 1.0f / sum;
        for (int l = lane; l < Lsz; l += 32) sc[w * Lsz + l] *= invs;
    }
    __syncthreads();

    for (int i = tid; i < Nsz * 64; i += 256) {
        int n = i >> 6, d = i & 63;
        const float* vp = Vbuf + bs * Lsz * Dsz + h * 64 + d;
        float acc = 0.0f;
        for (int l = 0; l < Lsz; ++l)
            acc += sc[n * Lsz + l] * vp[(long long)l * Dsz];
        attn_out[(bs * Nsz + n) * Dsz + h * 64 + d] = acc;
    }
}

// ---------------------------------------------------------------------------
__global__ __launch_bounds__(256)
void boost_prep_kernel(const float* __restrict__ g, const float* __restrict__ outp,
                       const float* __restrict__ nonseq, const float* __restrict__ w_qout,
                       const float* __restrict__ w_bin,
                       float* __restrict__ Qres, float* __restrict__ Qhat)
{
    __shared__ float red[256];
    const int idx = blockIdx.x;
    const int t = idx & 15;
    const int s = (idx >> 4) % Ssz;
    const int b = idx / (Ssz * Tsz);
    const int tid = threadIdx.x;

    float v0, v1;
    if (t < Nsz) {  // uniform per block
        const long long base = (((long long)b * Ssz + s) * Nsz + t) * Dsz;
        v0 = g[base + tid] + outp[base + tid];
        v1 = g[base + tid + 256] + outp[base + tid + 256];
        float inv = block_rms_inv(v0, v1, red);
        v0 = v0 * inv * w_qout[(long long)s * Dsz + tid];
        v1 = v1 * inv * w_qout[(long long)s * Dsz + tid + 256];
    } else {
        const long long base = (((long long)b * Ssz + s) * Msz + (t - Nsz)) * Dsz;
        v0 = nonseq[base + tid];
        v1 = nonseq[base + tid + 256];
    }
    float inv2 = block_rms_inv(v0, v1, red);
    v0 = v0 * inv2 * w_bin[(long long)s * Dsz + tid];
    v1 = v1 * inv2 * w_bin[(long long)s * Dsz + tid + 256];

    const long long rbase = (((long long)s * Tsz + t) * Bsz + b) * Dsz;
    Qres[rbase + tid]       = v0;
    Qres[rbase + tid + 256] = v1;
    // Q_hat[b,s,i, t*32+u] = Q_[b,s,t, i*32+u]
    const int d0 = tid, d1 = tid + 256;
    Qhat[(((long long)s * Tsz + (d0 >> 5)) * Bsz + b) * Dsz + t * 32 + (d0 & 31)] = v0;
    Qhat[(((long long)s * Tsz + (d1 >> 5)) * Bsz + b) * Dsz + t * 32 + (d1 & 31)] = v1;
}

__global__ __launch_bounds__(256)
void silu_kernel(float* __restrict__ x, long long n)
{
    long long i = (long long)blockIdx.x * blockDim.x + threadIdx.x;
    if (i < n) {
        float v = x[i];
        x[i] = v / (1.0f + __expf(-v));
    }
}

__global__ __launch_bounds__(256)
void final_norm_kernel(const float* __restrict__ out2, const float* __restrict__ Qres,
                       const float* __restrict__ w_bout,
                       float* __restrict__ o1, float* __restrict__ o2)
{
    __shared__ float red[256];
    const int idx = blockIdx.x;
    const int t = idx & 15;
    const int s = (idx >> 4) % Ssz;
    const int b = idx / (Ssz * Tsz);
    const int tid = threadIdx.x;

    const long long base = (((long long)s * Tsz + t) * Bsz + b) * Dsz;
    float v0 = out2[base + tid] + Qres[base + tid];
    float v1 = out2[base + tid + 256] + Qres[base + tid + 256];
    float inv = block_rms_inv(v0, v1, red);
    v0 = v0 * inv * w_bout[(long long)s * Dsz + tid];
    v1 = v1 * inv * w_bout[(long long)s * Dsz + tid + 256];

    float* dst;
    long long obase;
    if (t < Nsz) { dst = o1; obase = (((long long)b * Ssz + s) * Nsz + t) * Dsz; }
    else         { dst = o2; obase = (((long long)b * Ssz + s) * Msz + (t - Nsz)) * Dsz; }
    dst[obase + tid]       = v0;
    dst[obase + tid + 256] = v1;
}

// ---------------------------------------------------------------------------
extern "C" void kernel_launch(void* const* d_in, const int* in_sizes, int n_in,
                              void* d_out, int out_size, void* d_ws, size_t ws_size,
                              hipStream_t stream)
{
    const float* g       = (const float*)d_in[0];
    const float* nonseq  = (const float*)d_in[1];
    const float* seq     = (const float*)d_in[2];
    const unsigned char* msk = (const unsigned char*)d_in[3];
    const float* kproj   = (const float*)d_in[4];
    const float* vproj   = (const float*)d_in[5];
    const float* oproj   = (const float*)d_in[6];
    const float* b_o     = (const float*)d_in[7];
    const float* ffnW    = (const float*)d_in[8];
    const float* ffnb    = (const float*)d_in[9];
    const float* ffnoW   = (const float*)d_in[10];
    const float* ffnob   = (const float*)d_in[11];
    const float* w_qin   = (const float*)d_in[12];
    const float* w_kv    = (const float*)d_in[13];
    const float* w_qout  = (const float*)d_in[14];
    const float* w_bin   = (const float*)d_in[15];
    const float* w_bout  = (const float*)d_in[16];

    float* out1  = (float*)d_out;                                   // [B,S,N,D]
    float* out2p = out1 + (long long)Bsz * Ssz * Nsz * Dsz;         // [B,S,M,D]
    float* hkv   = out2p + (long long)Bsz * Ssz * Msz * Dsz;        // Hkv [B,S,L,D]

    float* w = (float*)d_ws;
    const long long SZ_KV = (long long)Bsz * Ssz * Lsz * Dsz;
    float* Kbuf   = w;
    float* Vbuf   = Kbuf + SZ_KV;
    float* Qn     = Vbuf + SZ_KV;
    float* attnO  = Qn + (long long)Bsz * Ssz * Nsz * Dsz;
    float* outp   = attnO + (long long)Bsz * Ssz * Nsz * Dsz;
    float* Qres   = outp + (long long)Bsz * Ssz * Nsz * Dsz;        // [s][t][B][D]
    float* Qhat   = Qres + (long long)Ssz * Tsz * Bsz * Dsz;        // [s][t][B][D]
    float* hidden = Qhat + (long long)Ssz * Tsz * Bsz * Dsz;        // [s][t][B][H]
    float* o2buf  = hidden + (long long)Ssz * Tsz * Bsz * Hsz;      // [s][t][B][D]

    const long long DD = (long long)Dsz * Dsz;

    rmsnorm_rows_kernel<<<Bsz * Ssz * Lsz, 256, 0, stream>>>(seq, w_kv, hkv, Lsz);
    rmsnorm_rows_kernel<<<Bsz * Ssz * Nsz, 256, 0, stream>>>(g, w_qin, Qn, Nsz);

    gemm_wmma_f32_kernel<<<dim3(Lsz / TILE_M, Dsz / TILE_N, Bsz * Ssz), 128, 0, stream>>>(
        hkv, DD, kproj, DD, Ssz, nullptr, 0, Kbuf, DD, Lsz, Dsz, Dsz);
    gemm_wmma_f32_kernel<<<dim3(Lsz / TILE_M, Dsz / TILE_N, Bsz * Ssz), 128, 0, stream>>>(
        hkv, DD, vproj, DD, Ssz, nullptr, 0, Vbuf, DD, Lsz, Dsz, Dsz);

    attention_kernel<<<dim3(8, Ssz, Bsz), 256, 0, stream>>>(Qn, Kbuf, Vbuf, msk, attnO);

    gemm_wmma_f32_kernel<<<dim3(1, Dsz / TILE_N, Bsz * Ssz), 128, 0, stream>>>(
        attnO, (long long)Nsz * Dsz, oproj, DD, Ssz, b_o, Dsz,
        outp, (long long)Nsz * Dsz, Nsz, Dsz, Dsz);

    boost_prep_kernel<<<Bsz * Ssz * Tsz, 256, 0, stream>>>(g, outp, nonseq, w_qout, w_bin,
                                                           Qres, Qhat);

    gemm_wmma_f32_kernel<<<dim3(1, Hsz / TILE_N, Ssz * Tsz), 128, 0, stream>>>(
        Qhat, (long long)Bsz * Dsz, ffnW, (long long)Dsz * Hsz, Ssz * Tsz,
        ffnb, Hsz, hidden, (long long)Bsz * Hsz, Bsz, Hsz, Dsz);

    {
        long long n = (long long)Ssz * Tsz * Bsz * Hsz;
        silu_kernel<<<(int)((n + 255) / 256), 256, 0, stream>>>(hidden, n);
    }

    gemm_wmma_f32_kernel<<<dim3(1, Dsz / TILE_N, Ssz * Tsz), 128, 0, stream>>>(
        hidden, (long long)Bsz * Hsz, ffnoW, (long long)Hsz * Dsz, Ssz * Tsz,
        ffnob, Dsz, o2buf, (long long)Bsz * Dsz, Bsz, Dsz, Hsz);

    final_norm_kernel<<<Bsz * Ssz * Tsz, 256, 0, stream>>>(o2buf, Qres, w_bout, out1, out2p);
}